// PGLoss_16252156248309
// MI455X (gfx1250) — compile-verified
//
#include <hip/hip_runtime.h>
#include <hip/hip_bf16.h>
#include <math.h>

#define B_    32
#define H_    160
#define W_    160
#define HW_   (H_*W_)
#define TCLBS 64
#define SEQN  64
#define NC    37
#define MAXT  50
#define LEXT  101          // 2*MAXT+1
#define EPS_  1e-6f
#define NEG_  -1e30f

typedef float v2f_ __attribute__((ext_vector_type(2)));
typedef float v4f_ __attribute__((ext_vector_type(4)));
typedef float v8f_ __attribute__((ext_vector_type(8)));

// Non-temporal 128-bit streaming load (single-touch data: don't pollute L2).
__device__ __forceinline__ v4f_ ld4nt(const float* p) {
  return __builtin_nontemporal_load((const v4f_*)p);
}

// ---------------------------------------------------------------------------
// Wave32 sum using the matrix unit: V_WMMA_F32_16X16X4_F32 with all-ones B.
// A layout (16x4 f32): vgpr0 lanes0-15 -> A[m][0], lanes16-31 -> A[m][2];
// vgpr1 = 0 -> A[m][1]=A[m][3]=0.  With B=ones: D[m][n] = x[m] + x[m+16].
// D layout: lane n<16 holds rows 0-7 in d[0..7]; lane n+16 holds rows 8-15.
// Per-lane sum of d[0..7] gives half-sums; shfl_xor(16) completes the total.
// ---------------------------------------------------------------------------
__device__ __forceinline__ float wave_sum32(float x) {
  v2f_ a; a[0] = x;   a[1] = 0.0f;
  v2f_ bm; bm[0] = 1.0f; bm[1] = 1.0f;           // all-ones 4x16 B
  v8f_ c = {0.f,0.f,0.f,0.f,0.f,0.f,0.f,0.f};
  v8f_ d = __builtin_amdgcn_wmma_f32_16x16x4_f32(
      /*neg_a=*/false, a, /*neg_b=*/false, bm,
      /*c_mod=*/(short)0, c, /*reuse_a=*/false, /*reuse_b=*/false);
  float s = d[0]+d[1]+d[2]+d[3]+d[4]+d[5]+d[6]+d[7];
  s += __shfl_xor(s, 16, 32);
  return s;                                       // full sum in every lane
}

// ---------------------------------------------------------------------------
// Fused streaming pass (memory-bound: 17 ch x 819200 px x 4B ~= 56 MB, one
// touch per byte). Each thread owns 4 contiguous pixels -> every channel is a
// single non-temporal global_load_b128. Computes, in one pass:
//   per-b: sum(p*g), sum(p*p), sum(g*g)     (dice,  p=f_score*mask, g=tcl*mask)
//   global: border_num, dir_num, sum(w)     (smooth-L1 numerators, w=tcl*mask)
// grid = (25 chunks of 1024 px, B images); 256 threads (8 wave32).
// ---------------------------------------------------------------------------
__global__ __launch_bounds__(256) void spatial_kernel(
    const float* __restrict__ f_score, const float* __restrict__ f_border,
    const float* __restrict__ f_direction, const float* __restrict__ tcl,
    const float* __restrict__ bmaps, const float* __restrict__ dmaps,
    const float* __restrict__ tmask, float* __restrict__ ws)
{
  const int b = blockIdx.y;
  const int p = blockIdx.x * 1024 + threadIdx.x * 4;   // 25*1024 == HW_
  const size_t b1 = (size_t)b * HW_;
  const size_t b2 = (size_t)b * 2 * HW_;
  const size_t b3 = (size_t)b * 3 * HW_;
  const size_t b4 = (size_t)b * 4 * HW_;
  const size_t b5 = (size_t)b * 5 * HW_;

  // 17 channels -> 17 b128 NT loads, issued up front so they clause together.
  const v4f_ sc = ld4nt(f_score + b1 + p);
  const v4f_ g  = ld4nt(tcl     + b1 + p);
  const v4f_ mk = ld4nt(tmask   + b1 + p);
  v4f_ fb[4], bmc[4], fd[2], dmc[2];
  #pragma unroll
  for (int c = 0; c < 4; ++c) {
    fb[c]  = ld4nt(f_border + b4 + (size_t)c * HW_ + p);
    bmc[c] = ld4nt(bmaps    + b5 + (size_t)c * HW_ + p);
  }
  const v4f_ lnB = ld4nt(bmaps + b5 + 4 * HW_ + p);    // l_norm (border ch 4)
  #pragma unroll
  for (int c = 0; c < 2; ++c) {
    fd[c]  = ld4nt(f_direction + b2 + (size_t)c * HW_ + p);
    dmc[c] = ld4nt(dmaps       + b3 + (size_t)c * HW_ + p);
  }
  const v4f_ lnD = ld4nt(dmaps + b3 + 2 * HW_ + p);    // l_norm (dir ch 2)

  float pg = 0.f, pp = 0.f, gg = 0.f, bn = 0.f, dn = 0.f, wsum = 0.f;
  #pragma unroll
  for (int j = 0; j < 4; ++j) {
    const float pmv = sc[j] * mk[j], gmv = g[j] * mk[j];
    pg += pmv * gmv;  pp += pmv * pmv;  gg += gmv * gmv;
    wsum += gmv;                                        // w = l_score*l_mask
    float accB = 0.f;
    #pragma unroll
    for (int c = 0; c < 4; ++c) {
      const float ad = fabsf(bmc[c][j] - fb[c][j]);
      accB += (ad < 1.f) ? 0.5f * ad * ad : (ad - 0.5f);
    }
    bn += lnB[j] * accB * gmv;
    float accD = 0.f;
    #pragma unroll
    for (int c = 0; c < 2; ++c) {
      const float ad = fabsf(dmc[c][j] - fd[c][j]);
      accD += (ad < 1.f) ? 0.5f * ad * ad : (ad - 0.5f);
    }
    dn += lnD[j] * accD * gmv;
  }

  // WMMA-based wave reduction, then cross-wave combine in LDS.
  float vals[6] = {pg, pp, gg, bn, dn, wsum};
  __shared__ float smem[8 * 6];
  const int lane = threadIdx.x & 31, wave = threadIdx.x >> 5;
  #pragma unroll
  for (int k = 0; k < 6; ++k) {
    const float s = wave_sum32(vals[k]);
    if (lane == 0) smem[wave * 6 + k] = s;
  }
  __syncthreads();
  if (threadIdx.x < 6) {
    float s = 0.f;
    #pragma unroll
    for (int w2 = 0; w2 < 8; ++w2) s += smem[w2 * 6 + threadIdx.x];
    const int k = threadIdx.x;
    if (k < 3) atomicAdd(&ws[b * 3 + k], s);      // per-b dice sums
    else       atomicAdd(&ws[96 + (k - 3)], s);   // 96:bn 97:dn 98:wsum
  }
}

// ---------------------------------------------------------------------------
// CTC: one block per batch item. Gather 37 logits per time step straight from
// NCHW f_char (4096x37 gathered reads instead of streaming the 121 MB tensor),
// mask, log-softmax in LDS, then the 64-step / 101-state alpha recursion with
// ping-pong LDS buffers (one barrier per time step).
// ---------------------------------------------------------------------------
__global__ __launch_bounds__(128) void ctc_kernel(
    const float* __restrict__ f_char, const float* __restrict__ pos_mask,
    const int* __restrict__ pos_list, const int* __restrict__ tcl_label,
    const int* __restrict__ label_t, float* __restrict__ ws)
{
  const int b = blockIdx.x, tid = threadIdx.x;
  __shared__ float lp[SEQN][NC + 3];              // 64 x 40 (bank spread)
  __shared__ float abuf[2][LEXT + 2];             // [0..1] = NEG guard cells
  __shared__ int   extl[LEXT];
  __shared__ int   skipl[LEXT];

  // Gather + pos_mask blend of logits.
  for (int idx = tid; idx < SEQN * NC; idx += 128) {
    const int t = idx / NC, c = idx - t * NC;
    const int i = b * SEQN + t;
    const int pb = pos_list[3 * i + 0];
    const int ph = pos_list[3 * i + 1];
    const int pw = pos_list[3 * i + 2];
    const float pm = pos_mask[i];
    const float v  = f_char[(((size_t)pb * NC) + c) * HW_ + (size_t)ph * W_ + pw];
    lp[t][c] = (c == NC - 1) ? (v * pm + (1.f - pm) * 20.f)
                             : (v * pm - (1.f - pm) * 20.f);
  }
  __syncthreads();

  // log_softmax over classes; extended-label table; buffer init.
  if (tid < SEQN) {
    float m = -INFINITY;
    for (int c = 0; c < NC; ++c) m = fmaxf(m, lp[tid][c]);
    float s = 0.f;
    for (int c = 0; c < NC; ++c) s += expf(lp[tid][c] - m);
    const float lse = m + logf(s);
    for (int c = 0; c < NC; ++c) lp[tid][c] -= lse;
  }
  if (tid < LEXT)     extl[tid] = (tid & 1) ? tcl_label[b * MAXT + (tid >> 1)] : (NC - 1);
  if (tid < LEXT + 2) { abuf[0][tid] = NEG_; abuf[1][tid] = NEG_; }
  __syncthreads();

  if (tid < LEXT) skipl[tid] = (tid >= 2 && extl[tid] != extl[tid - 2]) ? 1 : 0;
  if (tid < 2)    abuf[0][2 + tid] = lp[0][extl[tid]];   // alpha0
  __syncthreads();

  for (int t = 1; t < SEQN; ++t) {
    const int cur = t & 1, prev = cur ^ 1;
    if (tid < LEXT) {
      const float a  = abuf[prev][2 + tid];
      const float a1 = abuf[prev][1 + tid];
      const float a2 = skipl[tid] ? abuf[prev][tid] : NEG_;
      const float m  = fmaxf(a, fmaxf(a1, a2));
      const float s  = expf(a - m) + expf(a1 - m) + expf(a2 - m);
      abuf[cur][2 + tid] = m + logf(s) + lp[t][extl[tid]];
    }
    __syncthreads();
  }

  if (tid == 0) {
    const int tl = label_t[b];
    const int fin = (SEQN - 1) & 1;
    const float aL = abuf[fin][2 + 2 * tl];
    const float aP = (tl > 0) ? abuf[fin][2 + 2 * tl - 1] : NEG_;
    const float m  = fmaxf(aL, aP);
    const float nll = -(m + log1pf(expf(fminf(aL, aP) - m)));
    atomicAdd(&ws[99], nll);
  }
}

// ---------------------------------------------------------------------------
// Finalize: dice mean via WMMA wave reduction (one lane per batch item).
// ---------------------------------------------------------------------------
__global__ __launch_bounds__(32) void finalize_kernel(
    const float* __restrict__ ws, float* __restrict__ out)
{
  const int tid = threadIdx.x;                    // 32 lanes == B_
  const float pg = ws[tid * 3 + 0], pp = ws[tid * 3 + 1], gg = ws[tid * 3 + 2];
  const float r  = 2.f * pg / (pp + gg + 2.f * EPS_);
  const float tot = wave_sum32(r);
  if (tid == 0) {
    const float score_loss     = 1.f - tot / (float)B_;
    const float border_loss    = ws[96] / (4.f * ws[98] + 1e-5f);
    const float direction_loss = ws[97] / (2.f * ws[98] + 1e-5f);
    const float ctc_loss       = ws[99] / (float)TCLBS;
    out[0] = score_loss + border_loss + direction_loss + 5.f * ctc_loss;
    out[1] = score_loss;
    out[2] = border_loss;
    out[3] = direction_loss;
    out[4] = ctc_loss;
  }
}

__global__ __launch_bounds__(128) void init_ws(float* __restrict__ ws) {
  if (threadIdx.x < 100) ws[threadIdx.x] = 0.f;
}

extern "C" void kernel_launch(void* const* d_in, const int* in_sizes, int n_in,
                              void* d_out, int out_size, void* d_ws, size_t ws_size,
                              hipStream_t stream) {
  const float* f_score        = (const float*)d_in[0];
  const float* f_border       = (const float*)d_in[1];
  const float* f_direction    = (const float*)d_in[2];
  const float* f_char         = (const float*)d_in[3];
  const float* tcl_maps       = (const float*)d_in[4];
  const float* border_maps    = (const float*)d_in[5];
  const float* direction_maps = (const float*)d_in[6];
  const float* training_masks = (const float*)d_in[7];
  const float* pos_mask       = (const float*)d_in[8];
  const int*   pos_list       = (const int*)d_in[9];
  const int*   tcl_label      = (const int*)d_in[10];
  const int*   label_t        = (const int*)d_in[11];
  float* ws  = (float*)d_ws;
  float* out = (float*)d_out;

  hipLaunchKernelGGL(init_ws, dim3(1), dim3(128), 0, stream, ws);
  hipLaunchKernelGGL(spatial_kernel, dim3(25, B_), dim3(256), 0, stream,
                     f_score, f_border, f_direction, tcl_maps, border_maps,
                     direction_maps, training_masks, ws);
  hipLaunchKernelGGL(ctc_kernel, dim3(TCLBS), dim3(128), 0, stream,
                     f_char, pos_mask, pos_list, tcl_label, label_t, ws);
  hipLaunchKernelGGL(finalize_kernel, dim3(1), dim3(32), 0, stream, ws, out);
}